// VectorQuantizer_78417512890527
// MI455X (gfx1250) — compile-verified
//
#include <hip/hip_runtime.h>

// VectorQuantizer for MI455X (gfx1250, wave32, WMMA + async global->LDS DMA).
// dist(n,k) = |x_n|^2 - 2 x_n.c_k + |c_k|^2 ; argmin_k is invariant to |x_n|^2,
// so we minimize s = |c_k|^2 - 2*(x_n . c_k), with dot products computed by
// V_WMMA_F32_16X16X4_F32 (exact fp32, matching Precision.HIGHEST reference).
//
// Steady state: codebook tiles are staged into double-buffered LDS with
// GLOBAL_LOAD_ASYNC_TO_LDS_B128 (ASYNCcnt), zero staging VGPRs. To keep alias
// analysis honest about the DMA-written LDS (the asm only sees integers):
//  (a) the DMA LDS addresses are derived from ptrtoint of the shared array
//      (captures/escapes it, so the "memory"-clobbered asm may write it), and
//  (b) tile 0 is staged with ordinary loads+LDS stores in the prologue, so the
//      shared object has visible stores (defeats the loads-from-never-stored-
//      global -> undef fold that hit round 2). No volatile anywhere (volatile
//      lowers to serialized SYS-scope flat loads on HIP -- round 3 regression).

typedef __attribute__((ext_vector_type(2))) float v2f;
typedef __attribute__((ext_vector_type(8))) float v8f;

#define N_ROWS   32768          // B*H*W
#define DIM      256            // D
#define KCB      8192           // codebook entries
#define COLT     16             // codebook columns per WMMA tile
#define NTILES   (KCB / COLT)   // 512
#define LDS_STRIDE   260        // floats per staged row (260%64==4 -> conflict-free ds reads)
#define LDS_STRIDE4  (LDS_STRIDE / 4)  // 65 float4s per row
#define WAVES_PER_BLOCK 8       // 256 threads
#define ROWS_PER_BLOCK  (WAVES_PER_BLOCK * 16)  // 128

// ---------------------------------------------------------------------------
// Kernel 1: |c_k|^2 for every codebook row. One wave32 per row.
// ---------------------------------------------------------------------------
__global__ __launch_bounds__(256)
void vq_cbnorm_kernel(const float* __restrict__ cb, float* __restrict__ b2) {
    const int w    = (blockIdx.x * blockDim.x + threadIdx.x) >> 5;  // wave id == row
    const int lane = threadIdx.x & 31;
    if (w >= KCB) return;
    const float4* row = (const float4*)(cb + (size_t)w * DIM);
    float s = 0.0f;
#pragma unroll
    for (int i = 0; i < 2; ++i) {
        float4 v = row[lane + 32 * i];
        s += v.x * v.x + v.y * v.y + v.z * v.z + v.w * v.w;
    }
#pragma unroll
    for (int off = 16; off >= 1; off >>= 1)
        s += __shfl_xor(s, off, 32);
    if (lane == 0) b2[w] = s;
}

// ---------------------------------------------------------------------------
// Async copy of one float4 from cb[goff_bytes] to the LDS byte address in lds_bytes (per lane).
// GVS mode: global address is the SGPR base plus a 32-bit VGPR byte offset; vdst holds the LDS byte address.
// Tracked by ASYNCcnt.
// ---------------------------------------------------------------------------
__device__ __forceinline__
void async_cp_b128(unsigned lds_bytes, unsigned goff_bytes, const float* base) {
    asm volatile("global_load_async_to_lds_b128 %0, %1, %2"
                 :: "v"(lds_bytes), "v"(goff_bytes), "s"(base)
                 : "memory");
}
__device__ __forceinline__ void wait_async0() {
    asm volatile("s_wait_asynccnt 0x0" ::: "memory");
}

// ---------------------------------------------------------------------------
// Kernel 2: fused GEMM (WMMA f32 16x16x4) + argmin + gather.
// 256 threads = 8 waves; each wave owns 16 query rows (A panel in 128 VGPRs,
// reused across all 512 codebook tiles). All 8 waves share each codebook tile
// through async-DMA'd double-buffered LDS.
// ---------------------------------------------------------------------------
__global__ __launch_bounds__(256, 1)
void vq_main_kernel(const float* __restrict__ x,
                    const float* __restrict__ cb,
                    const float* __restrict__ b2,
                    float* __restrict__ outq,
                    float* __restrict__ outi) {
    __shared__ float4 lds4[2 * 16 * LDS_STRIDE4];   // 33,280 bytes, only LDS object
    float* lds = (float*)lds4;
    // The low 32 bits of the generic LDS pointer are the LDS byte address (aperture rule).
    // The ptrtoint also captures lds4, so the async-asm writes are honored by AA.
    const unsigned lds_base_bytes = (unsigned)(uintptr_t)lds4;

    const int lane = threadIdx.x & 31;
    const int wave = threadIdx.x >> 5;
    const int row_base = (blockIdx.x * WAVES_PER_BLOCK + wave) * 16;
    const int m    = lane & 15;           // row (A) / col (B,C) within tile
    const int koff = (lane >> 4) << 1;    // K sub: lanes 0-15 -> {0,1}, 16-31 -> {2,3}

    // Per-thread DMA slots (float4 e = tid + i*256: row=e>>6, col4=e&63).
    unsigned lslot[4], gslot[4];
#pragma unroll
    for (int i = 0; i < 4; ++i) {
        const unsigned e = (unsigned)threadIdx.x + i * 256u;
        lslot[i] = lds_base_bytes + ((e >> 6) * LDS_STRIDE4 + (e & 63u)) * 16u;
        gslot[i] = e * 16u;                                 // global bytes in tile
    }

    // ---- Prologue: stage tile 0 into buffer 0 with ordinary loads + stores
    // (real, visible stores to the shared object; regs die before the loop).
    {
        const float4* cb4 = (const float4*)cb;
        float4 sreg[4];
#pragma unroll
        for (int i = 0; i < 4; ++i) {
            const int e = threadIdx.x + i * 256;
            sreg[i] = cb4[e];
        }
#pragma unroll
        for (int i = 0; i < 4; ++i) {
            const int e = threadIdx.x + i * 256;
            lds4[(e >> 6) * LDS_STRIDE4 + (e & 63)] = sreg[i];
        }
    }

    // ---- Preload this wave's 16x256 A panel as 64 WMMA fragments (128 VGPRs).
    v2f a[64];
    {
        const float* xrow = x + (size_t)(row_base + m) * DIM + koff;
#pragma unroll
        for (int t = 0; t < 64; ++t)
            a[t] = *(const v2f*)(xrow + 4 * t);
    }

    float mind[8];
    int   midx[8];
#pragma unroll
    for (int r = 0; r < 8; ++r) { mind[r] = 3.4e38f; midx[r] = 0; }

    __syncthreads();

    // ---- Main loop over 512 codebook column tiles (DMA double-buffered).
    for (int tile = 0; tile < NTILES; ++tile) {
        const int cur = tile & 1;

        // DMA next tile into the other buffer while we compute on this one.
        if (tile + 1 < NTILES) {
            const unsigned gbase = (unsigned)(tile + 1) * (COLT * DIM * 4u);
            const unsigned lbase = (unsigned)(cur ^ 1) * (16u * LDS_STRIDE4 * 16u);
#pragma unroll
            for (int i = 0; i < 4; ++i)
                async_cp_b128(lbase + lslot[i], gbase + gslot[i], cb);
        }

        const float b2v = b2[tile * COLT + m];
        const float* Bp = lds + cur * (16 * LDS_STRIDE) + m * LDS_STRIDE + koff;

        // 64 WMMAs (16x16x4 f32) over D=256, 4 accumulator chains for ILP.
        v8f c0 = {}, c1 = {}, c2 = {}, c3 = {};
#pragma unroll
        for (int t = 0; t < 64; t += 4) {
            v2f bf0 = *(const v2f*)(Bp + 4 * (t + 0));
            v2f bf1 = *(const v2f*)(Bp + 4 * (t + 1));
            v2f bf2 = *(const v2f*)(Bp + 4 * (t + 2));
            v2f bf3 = *(const v2f*)(Bp + 4 * (t + 3));
            c0 = __builtin_amdgcn_wmma_f32_16x16x4_f32(false, a[t + 0], false, bf0,
                                                       (short)0, c0, false, false);
            c1 = __builtin_amdgcn_wmma_f32_16x16x4_f32(false, a[t + 1], false, bf1,
                                                       (short)0, c1, false, false);
            c2 = __builtin_amdgcn_wmma_f32_16x16x4_f32(false, a[t + 2], false, bf2,
                                                       (short)0, c2, false, false);
            c3 = __builtin_amdgcn_wmma_f32_16x16x4_f32(false, a[t + 3], false, bf3,
                                                       (short)0, c3, false, false);
        }
        v8f cs = (c0 + c1) + (c2 + c3);

        // Running argmin: C layout -> VGPR r holds (row_base+r, col) on lanes 0-15
        // and (row_base+r+8, col) on lanes 16-31, col = tile*16 + (lane&15).
        const int col = tile * COLT + m;
#pragma unroll
        for (int r = 0; r < 8; ++r) {
            float s = fmaf(-2.0f, cs[r], b2v);
            if (s < mind[r]) { mind[r] = s; midx[r] = col; }  // strict < keeps lowest idx
        }

        wait_async0();      // our DMA into the other buffer has landed
        __syncthreads();    // everyone done reading cur + next buffer visible
    }

    // ---- Cross-lane argmin within each 16-lane half (xor<16 stays in-half).
#pragma unroll
    for (int r = 0; r < 8; ++r) {
        float d = mind[r];
        int   ix = midx[r];
#pragma unroll
        for (int off = 8; off >= 1; off >>= 1) {
            float od = __shfl_xor(d, off, 32);
            int   oi = __shfl_xor(ix, off, 32);
            if (od < d || (od == d && oi < ix)) { d = od; ix = oi; }
        }
        mind[r] = d;
        midx[r] = ix;
    }

    // ---- Outputs: wave-cooperative gather of winning codebook rows + indices.
#pragma unroll
    for (int rr = 0; rr < 16; ++rr) {
        const int srcLane = (rr < 8) ? 0 : 16;
        const int idx = __shfl(midx[rr & 7], srcLane, 32);
        const int row = row_base + rr;
        const float4* src = (const float4*)(cb + (size_t)idx * DIM);
        float4* dst = (float4*)(outq + (size_t)row * DIM);
        dst[lane]      = src[lane];
        dst[lane + 32] = src[lane + 32];
        if (lane == 0) outi[row] = (float)idx;   // K=8192 exact in fp32
    }
}

// ---------------------------------------------------------------------------
extern "C" void kernel_launch(void* const* d_in, const int* in_sizes, int n_in,
                              void* d_out, int out_size, void* d_ws, size_t ws_size,
                              hipStream_t stream) {
    const float* x  = (const float*)d_in[0];   // (32,32,32,256) f32
    const float* cb = (const float*)d_in[1];   // (8192,256) f32
    float* outq = (float*)d_out;                         // quantized: N*D floats
    float* outi = outq + (size_t)N_ROWS * DIM;           // indices: N floats
    float* b2   = (float*)d_ws;                          // 8192 floats (32 KB scratch)

    vq_cbnorm_kernel<<<KCB / WAVES_PER_BLOCK, 256, 0, stream>>>(cb, b2);
    vq_main_kernel<<<N_ROWS / ROWS_PER_BLOCK, 256, 0, stream>>>(x, cb, b2, outq, outi);
}